// DeformRoIPooling_41463614276037
// MI455X (gfx1250) — compile-verified
//
#include <hip/hip_runtime.h>
#include <stdint.h>

// ---- problem constants (from the reference) ----
#define SPATIAL_SCALE 0.0625f
#define TRANS_STD     0.1f
#define PDIM          7      // OUT_SIZE == PART_SIZE == 7
#define SSAMP         4      // SAMPLE_PER_PART
#define CDIM          256    // OUTPUT_DIM (GROUP_SIZE == 1)
#define HDIM          128
#define WDIM          128
#define BATCH         2
#define NROI          128
#define HWDIM         (HDIM * WDIM)

// ------------------------------------------------------------------
// Kernel 1: NCHW -> NHWC transpose of the feature map.
// Load phase uses the CDNA5 async memory->LDS path:
//   global_load_async_to_lds_b128  (ASYNCcnt)  +  s_wait_asynccnt 0
// Per block: one 32(C) x 32(HW) tile. grid = (HW/32, C/32, B), 256 thr.
// ------------------------------------------------------------------
__global__ __launch_bounds__(256) void transpose_nchw_to_nhwc(
    const float* __restrict__ src, float* __restrict__ dst) {
  __shared__ float tile[32][33];  // +1 pad to dodge bank conflicts
  const int l   = threadIdx.x;          // 0..255
  const int hw0 = blockIdx.x * 32;      // spatial tile origin
  const int c0  = blockIdx.y * 32;      // channel tile origin
  const int bz  = blockIdx.z;           // batch

  const float* s = src + (size_t)bz * (CDIM * HWDIM);
  float*       d = dst + (size_t)bz * (CDIM * HWDIM);

  {
    const int crow = l >> 3;            // 0..31 channel row in tile
    const int hw4  = (l & 7) << 2;      // 0,4,...,28
    // per-lane: 16B memory -> 16B LDS, no VGPR round trip.
    // low 32 bits of a flat shared-memory pointer == LDS byte offset.
    const unsigned lds_dst = (unsigned)(uintptr_t)(&tile[crow][hw4]);
    const float*   gsrc    = s + (size_t)(c0 + crow) * HWDIM + hw0 + hw4;
    asm volatile("global_load_async_to_lds_b128 %0, %1, off"
                 :
                 : "v"(lds_dst), "v"(gsrc)
                 : "memory");
  }
  asm volatile("s_wait_asynccnt 0x0" ::: "memory");
  __syncthreads();
  {
    const int hwrow = l >> 3;           // 0..31 spatial row in tile
    const int c4    = (l & 7) << 2;     // 0,4,...,28
    float4 v;
    v.x = tile[c4 + 0][hwrow];
    v.y = tile[c4 + 1][hwrow];
    v.z = tile[c4 + 2][hwrow];
    v.w = tile[c4 + 3][hwrow];
    *(float4*)(d + (size_t)(hw0 + hwrow) * CDIM + c0 + c4) = v;
  }
}

// ---- shared ROI geometry helper (uniform per block -> SALU) ----
struct RoiGeom {
  int b;
  float wstart, hstart, subw, subh;
};

__device__ __forceinline__ RoiGeom roi_geometry(const float* __restrict__ rois,
                                                const float* __restrict__ offset,
                                                int n, int ph, int pw) {
  RoiGeom g;
  const float* roi = rois + n * 5;
  g.b = (int)roi[0];
  const float rsw = rintf(roi[1]) * SPATIAL_SCALE - 0.5f;
  const float rsh = rintf(roi[2]) * SPATIAL_SCALE - 0.5f;
  const float rew = (rintf(roi[3]) + 1.0f) * SPATIAL_SCALE - 0.5f;
  const float reh = (rintf(roi[4]) + 1.0f) * SPATIAL_SCALE - 0.5f;
  const float rw  = fmaxf(rew - rsw, 0.1f);
  const float rh  = fmaxf(reh - rsh, 0.1f);
  const float binw = rw / (float)PDIM;
  const float binh = rh / (float)PDIM;
  g.subw = binw / (float)SSAMP;
  g.subh = binh / (float)SSAMP;
  // part_h == ph, part_w == pw exactly (P == PART_SIZE; fp verified)
  const float tx = offset[((n * 2 + 0) * PDIM + ph) * PDIM + pw] * TRANS_STD;
  const float ty = offset[((n * 2 + 1) * PDIM + ph) * PDIM + pw] * TRANS_STD;
  g.wstart = (float)pw * binw + rsw + tx * rw;
  g.hstart = (float)ph * binh + rsh + ty * rh;
  return g;
}

// ------------------------------------------------------------------
// Kernel 2a: deformable PS-RoI pooling on the NHWC copy.
// One block per (roi n, bin); 64 threads (2 wave32), 4 channels/thread.
// Every bilinear corner is one coalesced global_load_b128 per lane.
// ------------------------------------------------------------------
__global__ __launch_bounds__(64) void deform_psroi_pool_nhwc4(
    const float* __restrict__ data,
    const float* __restrict__ rois,
    const float* __restrict__ offset,
    float* __restrict__ out) {
  const int bin = blockIdx.x;           // 0..48
  const int n   = blockIdx.y;           // 0..127
  const int ph  = bin / PDIM;
  const int pw  = bin % PDIM;
  const int c4  = threadIdx.x << 2;     // 0,4,...,252

  const RoiGeom g = roi_geometry(rois, offset, n, ph, pw);
  const float* base = data + (size_t)g.b * HWDIM * CDIM + c4;

  // gfx1250 vmem prefetch (global_prefetch_b8)
  __builtin_prefetch((const void*)base, 0, 1);

  float4 ssum = {0.0f, 0.0f, 0.0f, 0.0f};
  float count = 0.0f;

#pragma unroll
  for (int iy = 0; iy < SSAMP; ++iy) {
    const float hval = g.hstart + (float)iy * g.subh;
#pragma unroll
    for (int ix = 0; ix < SSAMP; ++ix) {
      const float wval = g.wstart + (float)ix * g.subw;
      const bool valid = (wval >= -0.5f) & (wval <= (float)WDIM - 0.5f) &
                         (hval >= -0.5f) & (hval <= (float)HDIM - 0.5f);
      const float wc = fminf(fmaxf(wval, 0.0f), (float)(WDIM - 1));
      const float hc = fminf(fmaxf(hval, 0.0f), (float)(HDIM - 1));
      const float y0f = floorf(hc);
      const float x0f = floorf(wc);
      const float y1f = ceilf(hc);
      const float x1f = ceilf(wc);
      const float dy = hc - y0f;
      const float dx = wc - x0f;
      const int y0 = (int)y0f, y1 = (int)y1f;
      const int x0 = (int)x0f, x1 = (int)x1f;

      const float4 q00 = *(const float4*)(base + (size_t)(y0 * WDIM + x0) * CDIM);
      const float4 q01 = *(const float4*)(base + (size_t)(y0 * WDIM + x1) * CDIM);
      const float4 q10 = *(const float4*)(base + (size_t)(y1 * WDIM + x0) * CDIM);
      const float4 q11 = *(const float4*)(base + (size_t)(y1 * WDIM + x1) * CDIM);

      const float w00 = (1.0f - dy) * (1.0f - dx);
      const float w01 = (1.0f - dy) * dx;
      const float w10 = dy * (1.0f - dx);
      const float w11 = dy * dx;

      if (valid) {
        ssum.x += w00 * q00.x + w01 * q01.x + w10 * q10.x + w11 * q11.x;
        ssum.y += w00 * q00.y + w01 * q01.y + w10 * q10.y + w11 * q11.y;
        ssum.z += w00 * q00.z + w01 * q01.z + w10 * q10.z + w11 * q11.z;
        ssum.w += w00 * q00.w + w01 * q01.w + w10 * q10.w + w11 * q11.w;
        count  += 1.0f;
      }
    }
  }

  const float inv = (count > 0.0f) ? (1.0f / count) : 0.0f;  // exact /count below
  (void)inv;
  const size_t obase = ((size_t)n * CDIM + c4) * (PDIM * PDIM) + bin;
  out[obase + 0 * (PDIM * PDIM)] = (count > 0.0f) ? (ssum.x / count) : 0.0f;
  out[obase + 1 * (PDIM * PDIM)] = (count > 0.0f) ? (ssum.y / count) : 0.0f;
  out[obase + 2 * (PDIM * PDIM)] = (count > 0.0f) ? (ssum.z / count) : 0.0f;
  out[obase + 3 * (PDIM * PDIM)] = (count > 0.0f) ? (ssum.w / count) : 0.0f;
}

// ------------------------------------------------------------------
// Kernel 2b: scalar NCHW fallback (only if ws_size < 32MB).
// ------------------------------------------------------------------
__global__ __launch_bounds__(256) void deform_psroi_pool_nchw(
    const float* __restrict__ data,
    const float* __restrict__ rois,
    const float* __restrict__ offset,
    float* __restrict__ out) {
  const int bin = blockIdx.x;
  const int n   = blockIdx.y;
  const int ph  = bin / PDIM;
  const int pw  = bin % PDIM;
  const int c   = threadIdx.x;

  const RoiGeom g = roi_geometry(rois, offset, n, ph, pw);
  const float* base = data + (size_t)(g.b * CDIM + c) * HWDIM;

  float ssum = 0.0f, count = 0.0f;
#pragma unroll
  for (int iy = 0; iy < SSAMP; ++iy) {
    const float hval = g.hstart + (float)iy * g.subh;
#pragma unroll
    for (int ix = 0; ix < SSAMP; ++ix) {
      const float wval = g.wstart + (float)ix * g.subw;
      const bool valid = (wval >= -0.5f) & (wval <= (float)WDIM - 0.5f) &
                         (hval >= -0.5f) & (hval <= (float)HDIM - 0.5f);
      const float wc = fminf(fmaxf(wval, 0.0f), (float)(WDIM - 1));
      const float hc = fminf(fmaxf(hval, 0.0f), (float)(HDIM - 1));
      const float y0f = floorf(hc), x0f = floorf(wc);
      const float dy = hc - y0f, dx = wc - x0f;
      const int y0 = (int)y0f, y1 = (int)ceilf(hc);
      const int x0 = (int)x0f, x1 = (int)ceilf(wc);
      const float v00 = base[y0 * WDIM + x0];
      const float v01 = base[y0 * WDIM + x1];
      const float v10 = base[y1 * WDIM + x0];
      const float v11 = base[y1 * WDIM + x1];
      const float val = (1.0f - dy) * (1.0f - dx) * v00 + (1.0f - dy) * dx * v01 +
                        dy * (1.0f - dx) * v10 + dy * dx * v11;
      if (valid) { ssum += val; count += 1.0f; }
    }
  }
  out[(size_t)(n * CDIM + c) * (PDIM * PDIM) + bin] =
      (count > 0.0f) ? (ssum / count) : 0.0f;
}

// ------------------------------------------------------------------
extern "C" void kernel_launch(void* const* d_in, const int* in_sizes, int n_in,
                              void* d_out, int out_size, void* d_ws, size_t ws_size,
                              hipStream_t stream) {
  const float* data   = (const float*)d_in[0];
  const float* rois   = (const float*)d_in[1];
  const float* offset = (const float*)d_in[2];
  float*       out    = (float*)d_out;

  const size_t data_bytes = (size_t)BATCH * CDIM * HWDIM * sizeof(float);  // 32 MB
  const dim3 pgrid(PDIM * PDIM, NROI);   // 49 x 128 blocks

  if (ws_size >= data_bytes) {
    float* t = (float*)d_ws;
    const dim3 tgrid(HWDIM / 32, CDIM / 32, BATCH);  // 512 x 8 x 2
    transpose_nchw_to_nhwc<<<tgrid, dim3(256), 0, stream>>>(data, t);
    deform_psroi_pool_nhwc4<<<pgrid, dim3(64), 0, stream>>>(t, rois, offset, out);
  } else {
    deform_psroi_pool_nchw<<<pgrid, dim3(CDIM), 0, stream>>>(data, rois, offset, out);
  }
}